// FactorizedVectorQuantizer_56856777064846
// MI455X (gfx1250) — compile-verified
//
#include <hip/hip_runtime.h>

typedef __attribute__((ext_vector_type(2))) float v2f;
typedef __attribute__((ext_vector_type(8))) float v8f;

#define KS 256
#define KC 16
#define HALF 64
#define DD 128
#define WPB 4
#define TPB (WPB * 32)
#define ROWS_PER_WAVE 32

// ws32 layout: [0]=mse(float bits) [1]=n_valid [2..257]=hist_shape [258..273]=hist_color
#define WS_WORDS 274

__global__ void vq_zero_ws(unsigned int* __restrict__ w)
{
    int i = threadIdx.x;
    if (i < WS_WORDS) w[i] = 0u;
}

__global__ __launch_bounds__(TPB) void vq_main(
    const float* __restrict__ x, const unsigned char* __restrict__ vmask,
    const float* __restrict__ gs, const float* __restrict__ gc,
    const float* __restrict__ wS, const float* __restrict__ wC,
    float* __restrict__ outq, unsigned int* __restrict__ ws32)
{
    __shared__ __align__(16) float ldsWS[KS * HALF];   // 64 KB shape codebook
    __shared__ __align__(16) float ldsWC[KC * HALF];   //  4 KB color codebook
    __shared__ float ldsWnS[KS];
    __shared__ float ldsWnC[KC];
    __shared__ unsigned int ldsHS[KS];
    __shared__ unsigned int ldsHC[KC];
    __shared__ int ldsAmax[WPB][ROWS_PER_WAVE][2];     // [wave][row][0=shape,1=color]

    const int tid  = threadIdx.x;
    const int lane = tid & 31;
    const int wv   = tid >> 5;
    const int r    = lane & 15;   // N / row-in-tile index
    const int h    = lane >> 4;   // half-wave select

    // ---- stage codebooks to LDS (async DMA path when available), zero histograms ----
#if __has_builtin(__builtin_amdgcn_global_load_async_to_lds_b128)
    {
        typedef int v4i_g __attribute__((vector_size(16)));          // matches builtin param
        typedef __attribute__((address_space(1))) v4i_g gv4;
        typedef __attribute__((address_space(3))) v4i_g lv4;
        gv4* gsrc = (gv4*)wS;
        lv4* ldst = (lv4*)ldsWS;
        for (int i = tid; i < KS * HALF / 4; i += TPB)
            __builtin_amdgcn_global_load_async_to_lds_b128(gsrc + i, ldst + i, 0, 0);
        gv4* gsrcc = (gv4*)wC;
        lv4* ldstc = (lv4*)ldsWC;
        for (int i = tid; i < KC * HALF / 4; i += TPB)
            __builtin_amdgcn_global_load_async_to_lds_b128(gsrcc + i, ldstc + i, 0, 0);
#if __has_builtin(__builtin_amdgcn_s_wait_asynccnt)
        __builtin_amdgcn_s_wait_asynccnt(0);
#else
        asm volatile("s_wait_asynccnt 0x0" ::: "memory");
#endif
    }
#else
    {
        const float4* s4 = (const float4*)wS;
        float4* d4 = (float4*)ldsWS;
        for (int i = tid; i < KS * HALF / 4; i += TPB) d4[i] = s4[i];
        const float4* s4c = (const float4*)wC;
        float4* d4c = (float4*)ldsWC;
        for (int i = tid; i < KC * HALF / 4; i += TPB) d4c[i] = s4c[i];
    }
#endif
    for (int i = tid; i < KS; i += TPB) ldsHS[i] = 0u;
    for (int i = tid; i < KC; i += TPB) ldsHC[i] = 0u;
    __syncthreads();

    // ---- codebook row squared norms ----
    for (int i = tid; i < KS; i += TPB) {
        const float* w = ldsWS + i * HALF;
        float s = 0.f;
        #pragma unroll 8
        for (int c = 0; c < HALF; ++c) s += w[c] * w[c];
        ldsWnS[i] = s;
    }
    for (int i = tid; i < KC; i += TPB) {
        const float* w = ldsWC + i * HALF;
        float s = 0.f;
        #pragma unroll 8
        for (int c = 0; c < HALF; ++c) s += w[c] * w[c];
        ldsWnC[i] = s;
    }
    __syncthreads();

    const int row0 = (blockIdx.x * WPB + wv) * ROWS_PER_WAVE;

    // ================= shape codebook pass (2 row-tiles share B) =================
    {
        const float* xrow0 = x + (size_t)(row0 + r) * DD;        // rows row0..row0+15
        const float* xrow1 = x + (size_t)(row0 + 16 + r) * DD;   // rows row0+16..row0+31
        v2f a0[16], a1[16];
        #pragma unroll
        for (int kk = 0; kk < 16; ++kk) {
            a0[kk].x = xrow0[4 * kk + 2 * h + 0];
            a0[kk].y = xrow0[4 * kk + 2 * h + 1];
            a1[kk].x = xrow1[4 * kk + 2 * h + 0];
            a1[kk].y = xrow1[4 * kk + 2 * h + 1];
        }
        float bestv[2][8]; int besti[2][8];
        #pragma unroll
        for (int j = 0; j < 8; ++j) {
            bestv[0][j] = -3.402823466e38f; besti[0][j] = 0;
            bestv[1][j] = -3.402823466e38f; besti[1][j] = 0;
        }
        const float* gb0 = gs + (size_t)(row0 + 8 * h) * KS + r;
        const float* gb1 = gs + (size_t)(row0 + 16 + 8 * h) * KS + r;

        for (int nt = 0; nt < KS / 16; ++nt) {
            const float* wb = ldsWS + (nt * 16 + r) * HALF + 2 * h;
            v2f b[16];
            #pragma unroll
            for (int kk = 0; kk < 16; ++kk) { b[kk].x = wb[4 * kk]; b[kk].y = wb[4 * kk + 1]; }
            v8f acc0 = {}, acc1 = {};
            #pragma unroll
            for (int kk = 0; kk < 16; ++kk) {      // two independent WMMA chains
                acc0 = __builtin_amdgcn_wmma_f32_16x16x4_f32(
                        false, a0[kk], false, b[kk], (short)0, acc0, false, false);
                acc1 = __builtin_amdgcn_wmma_f32_16x16x4_f32(
                        false, a1[kk], false, b[kk], (short)0, acc1, false, false);
            }
            const float wn = ldsWnS[nt * 16 + r];
            const float* gp0 = gb0 + nt * 16;
            const float* gp1 = gb1 + nt * 16;
            __builtin_prefetch(gp0 + 16, 0, 0);
            __builtin_prefetch(gp1 + 16, 0, 0);
            #pragma unroll
            for (int j = 0; j < 8; ++j) {
                float s0 = gp0[(size_t)j * KS] + 2.0f * acc0[j] - wn;
                if (s0 > bestv[0][j]) { bestv[0][j] = s0; besti[0][j] = nt * 16 + r; }
                float s1 = gp1[(size_t)j * KS] + 2.0f * acc1[j] - wn;
                if (s1 > bestv[1][j]) { bestv[1][j] = s1; besti[1][j] = nt * 16 + r; }
            }
        }
        // argmax across the 16 lanes of each half-wave (ties -> lowest index)
        #pragma unroll
        for (int it = 0; it < 2; ++it) {
            #pragma unroll
            for (int j = 0; j < 8; ++j) {
                float bv = bestv[it][j]; int bi = besti[it][j];
                #pragma unroll
                for (int m = 1; m < 16; m <<= 1) {
                    float ov = __shfl_xor(bv, m, 32);
                    int   oi = __shfl_xor(bi, m, 32);
                    if (ov > bv || (ov == bv && oi < bi)) { bv = ov; bi = oi; }
                }
                if (r == 0) ldsAmax[wv][it * 16 + 8 * h + j][0] = bi;
            }
        }
    }

    // ================= color codebook pass =================
    {
        const float* xrow0 = x + (size_t)(row0 + r) * DD + HALF;
        const float* xrow1 = x + (size_t)(row0 + 16 + r) * DD + HALF;
        v2f a0[16], a1[16];
        #pragma unroll
        for (int kk = 0; kk < 16; ++kk) {
            a0[kk].x = xrow0[4 * kk + 2 * h + 0];
            a0[kk].y = xrow0[4 * kk + 2 * h + 1];
            a1[kk].x = xrow1[4 * kk + 2 * h + 0];
            a1[kk].y = xrow1[4 * kk + 2 * h + 1];
        }
        const float* wb = ldsWC + r * HALF + 2 * h;
        v2f b[16];
        #pragma unroll
        for (int kk = 0; kk < 16; ++kk) { b[kk].x = wb[4 * kk]; b[kk].y = wb[4 * kk + 1]; }
        v8f acc0 = {}, acc1 = {};
        #pragma unroll
        for (int kk = 0; kk < 16; ++kk) {
            acc0 = __builtin_amdgcn_wmma_f32_16x16x4_f32(
                    false, a0[kk], false, b[kk], (short)0, acc0, false, false);
            acc1 = __builtin_amdgcn_wmma_f32_16x16x4_f32(
                    false, a1[kk], false, b[kk], (short)0, acc1, false, false);
        }
        const float wn = ldsWnC[r];
        const float* gp0 = gc + (size_t)(row0 + 8 * h) * KC + r;
        const float* gp1 = gc + (size_t)(row0 + 16 + 8 * h) * KC + r;
        #pragma unroll
        for (int it = 0; it < 2; ++it) {
            const float* gp = (it == 0) ? gp0 : gp1;
            const v8f acc = (it == 0) ? acc0 : acc1;
            #pragma unroll
            for (int j = 0; j < 8; ++j) {
                float bv = gp[(size_t)j * KC] + 2.0f * acc[j] - wn;
                int   bi = r;
                #pragma unroll
                for (int m = 1; m < 16; m <<= 1) {
                    float ov = __shfl_xor(bv, m, 32);
                    int   oi = __shfl_xor(bi, m, 32);
                    if (ov > bv || (ov == bv && oi < bi)) { bv = ov; bi = oi; }
                }
                if (r == 0) ldsAmax[wv][it * 16 + 8 * h + j][1] = bi;
            }
        }
    }

    __syncthreads();

    // ====== gather codebook rows, write quantized, masked MSE, histograms ======
    {
        float errsum = 0.f;
        int   nvc    = 0;
        #pragma unroll
        for (int it = 0; it < 2; ++it) {
            const int rr  = it * 16 + r;
            const int row = row0 + rr;                 // lane pair (r, r+16) covers one row
            const int idx = ldsAmax[wv][rr][h];
            const float* src = (h == 0) ? (ldsWS + idx * HALF) : (ldsWC + idx * HALF);
            const float* xr  = x    + (size_t)row * DD + HALF * h;
            float*       qr  = outq + (size_t)row * DD + HALF * h;
            const bool valid = vmask[row] != 0;

            float e = 0.f;
            #pragma unroll
            for (int c = 0; c < HALF; c += 4) {
                float4 w4 = *(const float4*)(src + c);      // ds_load_b128
                float4 x4 = *(const float4*)(xr + c);
                *(float4*)(qr + c) = w4;                    // global_store_b128
                float d0 = w4.x - x4.x, d1 = w4.y - x4.y, d2 = w4.z - x4.z, d3 = w4.w - x4.w;
                e += d0 * d0 + d1 * d1 + d2 * d2 + d3 * d3;
            }
            if (valid) {
                errsum += e;
                if (h == 0) { nvc++; atomicAdd(&ldsHS[idx], 1u); }
                else        { atomicAdd(&ldsHC[idx], 1u); }
            }
        }
        #pragma unroll
        for (int m = 16; m > 0; m >>= 1) {
            errsum += __shfl_xor(errsum, m, 32);
            nvc    += __shfl_xor(nvc, m, 32);
        }
        if (lane == 0) {
            atomicAdd(reinterpret_cast<float*>(ws32), errsum);
            atomicAdd(&ws32[1], (unsigned int)nvc);
        }
    }

    __syncthreads();
    for (int i = tid; i < KS; i += TPB) atomicAdd(&ws32[2 + i], ldsHS[i]);
    for (int i = tid; i < KC; i += TPB) atomicAdd(&ws32[2 + KS + i], ldsHC[i]);
}

__global__ void vq_finalize(const unsigned int* __restrict__ ws32, float* __restrict__ out3)
{
    __shared__ float red[256];
    const int t = threadIdx.x;
    const float nv = (float)ws32[1];

    float p = (float)ws32[2 + t] / nv;
    red[t] = p * logf(p + 1e-10f);
    __syncthreads();
    for (int s = 128; s > 0; s >>= 1) { if (t < s) red[t] += red[t + s]; __syncthreads(); }
    const float entS = red[0];
    __syncthreads();

    float pc = (t < KC) ? ((float)ws32[2 + KS + t] / nv) : 0.f;
    red[t] = (t < KC) ? pc * logf(pc + 1e-10f) : 0.f;
    __syncthreads();
    for (int s = 128; s > 0; s >>= 1) { if (t < s) red[t] += red[t + s]; __syncthreads(); }

    if (t == 0) {
        float mse = __uint_as_float(ws32[0]);
        out3[0] = 1.25f * mse / (nv * (float)DD);  // q_latent + COMMIT*e_latent, same fwd value
        out3[1] = expf(-entS);
        out3[2] = expf(-red[0]);
    }
}

extern "C" void kernel_launch(void* const* d_in, const int* in_sizes, int n_in,
                              void* d_out, int out_size, void* d_ws, size_t ws_size,
                              hipStream_t stream) {
    const float*         xin  = (const float*)d_in[0];
    const unsigned char* vm   = (const unsigned char*)d_in[1];   // jax bool -> 1 byte
    const float*         gs   = (const float*)d_in[2];
    const float*         gc   = (const float*)d_in[3];
    const float*         wSh  = (const float*)d_in[4];
    const float*         wCo  = (const float*)d_in[5];
    float*               out  = (float*)d_out;
    unsigned int*        ws32 = (unsigned int*)d_ws;

    const int Ntot   = in_sizes[0] / DD;                  // 131072
    const int blocks = Ntot / (ROWS_PER_WAVE * WPB);      // 1024

    vq_zero_ws<<<1, 512, 0, stream>>>(ws32);
    vq_main<<<blocks, TPB, 0, stream>>>(xin, vm, gs, gc, wSh, wCo, out, ws32);
    vq_finalize<<<1, 256, 0, stream>>>(ws32, out + (size_t)Ntot * DD);
}